// NodeShuffle_10651518894813
// MI455X (gfx1250) — compile-verified
//
#include <hip/hip_runtime.h>

typedef __attribute__((ext_vector_type(2))) float v2f;
typedef __attribute__((ext_vector_type(8))) float v8f;

#define NN   16384
#define CC   64
#define KNN  16
#define RR   4
#define COUT 256           // C*R
#define BIGF 3.0e38f
#define NITER (NN / 128)   // column tiles of 128 per workgroup

// ---------------------------------------------------------------------------
// Kernel 1: squared norms sq[i] = sum_c x[i][c]^2
// ---------------------------------------------------------------------------
__global__ __launch_bounds__(256) void sqnorm_kernel(const float* __restrict__ x,
                                                     float* __restrict__ sq) {
  int i = blockIdx.x * 256 + threadIdx.x;
  if (i < NN) {
    const float4* p = (const float4*)(x + (size_t)i * CC);
    float s = 0.0f;
#pragma unroll
    for (int q = 0; q < CC / 4; ++q) {
      float4 v = p[q];
      s += v.x * v.x + v.y * v.y + v.z * v.z + v.w * v.w;
    }
    sq[i] = s;
  }
}

// ---------------------------------------------------------------------------
// Kernel 2: fused  x @ x^T  (WMMA f32 16x16x4)  +  per-row top-16 selection.
// One workgroup = 8 waves = 16 query rows. Each wave owns a 16-column strip
// per 128-column tile; its 16x16 score tile is redistributed through a
// wave-private LDS slab (same-wave DS ordering, NO workgroup barrier in the
// main loop). B fragments are double-buffered in registers so the WMMA chain
// never stalls on memory.
// ---------------------------------------------------------------------------
__device__ __forceinline__ void load_btile(v2f b[16], const float* __restrict__ brow) {
#pragma unroll
  for (int kk = 0; kk < 16; ++kk)
    b[kk] = *(const v2f*)(brow + 4 * kk);
}

__global__ __launch_bounds__(256) void knn_kernel(const float* __restrict__ x,
                                                  const float* __restrict__ sq,
                                                  int* __restrict__ idxOut) {
  __shared__ float sc[8][16][17];  // per-wave 16x16 score tile (+pad)
  __shared__ float mD[16][256];    // merge buffers: 16 sorted lists per row
  __shared__ int   mI[16][256];

  const int lane = threadIdx.x & 31;
  const int wave = threadIdx.x >> 5;     // 0..7
  const int hi   = lane >> 4;            // 0/1
  const int lo   = lane & 15;            // 0..15
  const int q0   = blockIdx.x * 16;

  float (*scw)[17] = sc[wave];           // this wave's slab

  // Resident A-fragments for the 16 query rows.
  // Lane layout per ISA: v2f = ( x[q0+lo][4k+2*hi], x[q0+lo][4k+2*hi+1] )
  v2f afr[16];
  {
    const float* arow = x + (size_t)(q0 + lo) * CC + 2 * hi;
#pragma unroll
    for (int kk = 0; kk < 16; ++kk)
      afr[kk] = *(const v2f*)(arow + 4 * kk);
  }

  // Private sorted (ascending) top-16 list for partition (row=lo, wave, hi).
  float td[16];
  int   ti[16];
#pragma unroll
  for (int s = 0; s < 16; ++s) { td[s] = BIGF; ti[s] = 0; }

  // WMMA + score + wave-local select for one 16x16 strip of tile `it`.
  auto process = [&](int it, const v2f (&bfr)[16], float sqj) {
    v8f acc = {};
#pragma unroll
    for (int kk = 0; kk < 16; ++kk)
      acc = __builtin_amdgcn_wmma_f32_16x16x4_f32(
          false, afr[kk], false, bfr[kk], (short)0, acc, false, false);

    // acc[v] = dot(x[q0+m], x[colBase+n]), m = v + 8*hi, n = lo
#pragma unroll
    for (int v = 0; v < 8; ++v)
      scw[v + 8 * hi][lo] = fmaf(-2.0f, acc[v], sqj);

    // Same-wave DS ordering: stores above complete before loads below.
    asm volatile("s_wait_dscnt 0x0" ::: "memory");

    // Lane scans row (lo), columns hi*8 .. hi*8+7 of this wave's strip.
    const int colBase = it * 128 + wave * 16;
#pragma unroll
    for (int p = 0; p < 8; ++p) {
      const int lc   = hi * 8 + p;
      const int gcol = colBase + lc;
      float key = scw[lo][lc];
      if (gcol == q0 + lo) key = BIGF;        // exclude self
      if (key < td[15]) {                      // usually false -> cheap
        float kd = key; int ki = gcol;
#pragma unroll
        for (int s = 0; s < 16; ++s) {
          if (kd < td[s]) {
            float t0 = td[s]; td[s] = kd; kd = t0;
            int   t1 = ti[s]; ti[s] = ki; ki = t1;
          }
        }
      }
    }
  };

  // Software pipeline, 2x unrolled, B double-buffered in registers.
  v2f ba[16], bb[16];
  float sqa, sqb;
  {
    const int cb0 = wave * 16;
    load_btile(ba, x + (size_t)(cb0 + lo) * CC + 2 * hi);
    sqa = sq[cb0 + lo];
  }
  for (int it = 0; it < NITER; it += 2) {
    const int cb1 = (it + 1) * 128 + wave * 16;
    load_btile(bb, x + (size_t)(cb1 + lo) * CC + 2 * hi);
    sqb = sq[cb1 + lo];

    process(it, ba, sqa);

    if (it + 2 < NITER) {
      const int cb2 = (it + 2) * 128 + wave * 16;
      load_btile(ba, x + (size_t)(cb2 + lo) * CC + 2 * hi);
      sqa = sq[cb2 + lo];
    }

    process(it + 1, bb, sqb);
  }

  // Merge: each lane dumps its sorted list for (row=lo, partition=wave*2+hi);
  // 16 partitions x 16 entries = 256 candidates per row.
  {
    const int part = wave * 2 + hi;
#pragma unroll
    for (int s = 0; s < 16; ++s) {
      mD[lo][part * 16 + s] = td[s];
      mI[lo][part * 16 + s] = ti[s];
    }
  }
  __syncthreads();
  if (threadIdx.x < 16) {
    const int r = threadIdx.x;
    for (int o = 0; o < KNN; ++o) {
      float best = BIGF; int bp = 0;
      for (int p = 0; p < 256; ++p) {
        float v = mD[r][p];
        if (v < best) { best = v; bp = p; }
      }
      idxOut[(size_t)(q0 + r) * KNN + o] = mI[r][bp];
      mD[r][bp] = BIGF;
    }
  }
}

// ---------------------------------------------------------------------------
// Kernel 3: A = x @ (W_top - W_bot) + b ;  B = x @ W_bot   (WMMA f32 16x16x4)
// grid = (N/16, 2), 8 waves/block; wave w does col tile (blockIdx.y*8 + w).
// Loads fully batched before the dual WMMA chain.
// ---------------------------------------------------------------------------
__global__ __launch_bounds__(256) void featgemm_kernel(const float* __restrict__ x,
                                                       const float* __restrict__ W,
                                                       const float* __restrict__ bias,
                                                       float* __restrict__ Aout,
                                                       float* __restrict__ Bout) {
  const int lane = threadIdx.x & 31;
  const int wave = threadIdx.x >> 5;
  const int hi   = lane >> 4;
  const int lo   = lane & 15;
  const int q0   = blockIdx.x * 16;
  const int n    = (blockIdx.y * 8 + wave) * 16 + lo;   // output column

  v2f afr[16];
  {
    const float* arow = x + (size_t)(q0 + lo) * CC + 2 * hi;
#pragma unroll
    for (int kk = 0; kk < 16; ++kk)
      afr[kk] = *(const v2f*)(arow + 4 * kk);
  }

  // Batch all weight loads, form both B-fragment sets.
  v2f bcmb[16];   // W_top - W_bot
  v2f bbot[16];   // W_bot
#pragma unroll
  for (int kk = 0; kk < 16; ++kk) {
    const int k = 4 * kk + 2 * hi;
    const float wt0 = W[(size_t)k * COUT + n];
    const float wt1 = W[(size_t)(k + 1) * COUT + n];
    const float wb0 = W[(size_t)(k + CC) * COUT + n];
    const float wb1 = W[(size_t)(k + CC + 1) * COUT + n];
    v2f c; c.x = wt0 - wb0; c.y = wt1 - wb1;
    v2f d; d.x = wb0;       d.y = wb1;
    bcmb[kk] = c;
    bbot[kk] = d;
  }

  v8f acc1 = {};   // x @ (W_top - W_bot)
  v8f acc2 = {};   // x @ W_bot
#pragma unroll
  for (int kk = 0; kk < 16; ++kk) {
    acc1 = __builtin_amdgcn_wmma_f32_16x16x4_f32(
        false, afr[kk], false, bcmb[kk], (short)0, acc1, false, false);
    acc2 = __builtin_amdgcn_wmma_f32_16x16x4_f32(
        false, afr[kk], false, bbot[kk], (short)0, acc2, false, false);
  }

  const float bv = bias[n];
#pragma unroll
  for (int v = 0; v < 8; ++v) {
    const int m = v + 8 * hi;
    Aout[(size_t)(q0 + m) * COUT + n] = acc1[v] + bv;
    Bout[(size_t)(q0 + m) * COUT + n] = acc2[v];
  }
}

// ---------------------------------------------------------------------------
// Kernel 4: per node, max over 16 neighbors of (A[i] + B[j]), relu, and
// permuted store: y[(i*R + r)*C + c] = out[i, c*R + r].
// One block per node; thread t owns output feature t (0..255).
// ---------------------------------------------------------------------------
__global__ __launch_bounds__(256) void gathermax_kernel(const float* __restrict__ A,
                                                        const float* __restrict__ B,
                                                        const int* __restrict__ idx,
                                                        float* __restrict__ y) {
  const int i = blockIdx.x;
  const int t = threadIdx.x;
  const float a = A[(size_t)i * COUT + t];

  // Batch the index loads, then the 16 independent row gathers.
  int j[KNN];
#pragma unroll
  for (int k = 0; k < KNN; ++k) j[k] = idx[(size_t)i * KNN + k];

  float m = -BIGF;
#pragma unroll
  for (int k = 0; k < KNN; ++k)
    m = fmaxf(m, a + B[(size_t)j[k] * COUT + t]);

  m = fmaxf(m, 0.0f);                 // relu(max) == max(relu) (monotone)
  const int c = t >> 2;               // t / R
  const int r = t & 3;                // t % R
  y[((size_t)i * RR + r) * CC + c] = m;
}

// ---------------------------------------------------------------------------
extern "C" void kernel_launch(void* const* d_in, const int* in_sizes, int n_in,
                              void* d_out, int out_size, void* d_ws, size_t ws_size,
                              hipStream_t stream) {
  const float* x = (const float*)d_in[0];   // (16384, 64)
  const float* W = (const float*)d_in[1];   // (128, 256)
  const float* b = (const float*)d_in[2];   // (256,)
  float* y = (float*)d_out;                 // (65536, 64)

  char* ws = (char*)d_ws;
  float* sq   = (float*)ws;                                   //  64 KB
  int*   nidx = (int*)(ws + (size_t)NN * 4);                  //   1 MB
  float* Aout = (float*)(ws + (size_t)NN * 4 + (size_t)NN * KNN * 4);  // 16 MB
  float* Bout = Aout + (size_t)NN * COUT;                              // 16 MB

  sqnorm_kernel<<<NN / 256, 256, 0, stream>>>(x, sq);
  featgemm_kernel<<<dim3(NN / 16, 2), 256, 0, stream>>>(x, W, b, Aout, Bout);
  knn_kernel<<<NN / 16, 256, 0, stream>>>(x, sq, nidx);
  gathermax_kernel<<<NN, 256, 0, stream>>>(Aout, Bout, nidx, y);
}